// contrastive_45638322487804
// MI455X (gfx1250) — compile-verified
//
#include <hip/hip_runtime.h>
#include <hip/hip_bf16.h>
#include <math.h>

// ---- problem constants (from the reference) ----
#define N_PTS   16384
#define DIM     24
#define E_SIG_N 65536
#define E_RND_N 32768
#define KSEL    51        // K+1 nearest (incl. self)
#define CAP     256       // per-row candidate buffer (d^2 < 1 is sparse)
#define TILE    16
#define KNN_THREADS 256   // 8 wave32 per workgroup
#define KNN_WAVES   8

typedef __attribute__((ext_vector_type(2))) float v2f;
typedef __attribute__((ext_vector_type(8))) float v8f;

// ---------- squared norms: sq[i] = ||emb_i||^2 ----------
__global__ void sq_kernel(const float* __restrict__ emb, float* __restrict__ sq) {
  int i = blockIdx.x * blockDim.x + threadIdx.x;
  if (i < N_PTS) {
    float s = 0.f;
#pragma unroll
    for (int d = 0; d < DIM; ++d) { float v = emb[i * DIM + d]; s += v * v; }
    sq[i] = s;
  }
}

// ---------- edge losses (signal: mean d^2 ; random: masked hinge) ----------
__global__ void edge_kernel(const float* __restrict__ emb,
                            const int* __restrict__ edges,
                            const int* __restrict__ pid,
                            float* __restrict__ out,
                            int nE, float scale, int hinge) {
  int e = blockIdx.x * blockDim.x + threadIdx.x;
  float contrib = 0.f;
  if (e < nE) {
    int i = edges[e];
    int j = edges[nE + e];
    float d2 = 0.f;
#pragma unroll
    for (int d = 0; d < DIM; ++d) {
      float df = emb[i * DIM + d] - emb[j * DIM + d];
      d2 += df * df;
    }
    d2 += 1e-12f;
    if (hinge) {
      if (pid[i] != pid[j]) {
        float dd = sqrtf(d2);
        float h = fmaxf(1.0f - dd, 0.0f);
        contrib = h * h * scale;
      }
    } else {
      contrib = d2 * scale;    // ds*ds == d2 + 1e-12
    }
  }
  __shared__ float red[256];
  red[threadIdx.x] = contrib;
  __syncthreads();
  for (int s = blockDim.x >> 1; s > 0; s >>= 1) {
    if ((int)threadIdx.x < s) red[threadIdx.x] += red[threadIdx.x + s];
    __syncthreads();
  }
  if (threadIdx.x == 0) atomicAdd(out, red[0]);
}

// ---------- kNN hinge loss: 16-row tile per block, WMMA f32 16x16x4 ----------
__global__ __launch_bounds__(KNN_THREADS)
void knn_kernel(const float* __restrict__ emb,
                const float* __restrict__ sq,
                const int* __restrict__ pid,
                float* __restrict__ out) {
  __shared__ float s_d2[TILE * CAP];
  __shared__ int   s_col[TILE * CAP];
  __shared__ int   s_cnt[TILE];

  const int tid  = threadIdx.x;
  const int lane = tid & 31;
  const int wave = tid >> 5;
  const int row0 = blockIdx.x * TILE;

  if (tid < TILE) s_cnt[tid] = 0;
  __syncthreads();

  // WMMA f32 16x16x4 lane mapping (ISA 7.12.2):
  //   A: lanes 0-15 -> M=lane, K={k0,k0+1}; lanes 16-31 -> M=lane-16, K={k0+2,k0+3}
  //   B: same with N in place of M
  //   C: VGPR r, lane L -> M = r + 8*(L>>4), N = L & 15
  const int mA  = lane & 15;
  const int kHi = (lane >> 4) & 1;

  // Preload A fragments for this block's 16 rows (6 k-steps x 4 = 24 dims).
  v2f a[6];
#pragma unroll
  for (int kk = 0; kk < 6; ++kk) {
    const int k0 = kk * 4 + kHi * 2;                       // even -> 8B aligned
    const float2 t = *(const float2*)(emb + (row0 + mA) * DIM + k0);
    a[kk].x = t.x; a[kk].y = t.y;
  }
  // Preload the row squared-norms this lane will need for its C rows.
  float sqr[8];
#pragma unroll
  for (int r = 0; r < 8; ++r) sqr[r] = sq[row0 + r + 8 * (lane >> 4)];

  const int nTiles = N_PTS / TILE;
  for (int jt = wave; jt < nTiles; jt += KNN_WAVES) {       // uniform per wave
    const int col0 = jt * TILE;
    v8f c = {};
#pragma unroll
    for (int kk = 0; kk < 6; ++kk) {
      const int k0 = kk * 4 + kHi * 2;
      const float2 t = *(const float2*)(emb + (col0 + mA) * DIM + k0);
      v2f b; b.x = t.x; b.y = t.y;
      // D = A x B + C  (16x16 dot-product tile, f32 exact)
      c = __builtin_amdgcn_wmma_f32_16x16x4_f32(
              /*neg_a=*/false, a[kk], /*neg_b=*/false, b,
              /*c_mod=*/(short)0, c, /*reuse_a=*/false, /*reuse_b=*/false);
    }
    const int   col = col0 + (lane & 15);
    const float sqc = sq[col];

    // Compute all 8 distances, then guard the rare push path with ONE
    // compare+branch (single exec-mask toggle in the common empty case).
    float d2v[8];
    float dmin = 1e30f;
#pragma unroll
    for (int r = 0; r < 8; ++r) {
      d2v[r] = sqr[r] + sqc - 2.0f * c[r];
      dmin = fminf(dmin, d2v[r]);
    }
    if (dmin < 1.0f) {                                      // slow path (rare)
#pragma unroll
      for (int r = 0; r < 8; ++r) {
        if (d2v[r] < 1.0f) {                                // hinge/radius support
          const int M = r + 8 * (lane >> 4);
          int slot = atomicAdd(&s_cnt[M], 1);
          if (slot < CAP) { s_d2[M * CAP + slot] = d2v[r]; s_col[M * CAP + slot] = col; }
        }
      }
    }
  }
  __syncthreads();

  // Per-row tail: select the 51 smallest candidates, apply masks + hinge.
  if (tid < TILE) {
    const int row = row0 + tid;
    const int cnt = min(s_cnt[tid], CAP);
    float best[KSEL]; int bcol[KSEL]; int nb = 0;
    for (int iC = 0; iC < cnt; ++iC) {
      const float d2 = s_d2[tid * CAP + iC];
      const int   cc = s_col[tid * CAP + iC];
      int pos;
      if (nb < KSEL)                 pos = nb++;
      else if (d2 < best[KSEL - 1])  pos = KSEL - 1;
      else                           continue;
      while (pos > 0 && best[pos - 1] > d2) {
        best[pos] = best[pos - 1]; bcol[pos] = bcol[pos - 1]; --pos;
      }
      best[pos] = d2; bcol[pos] = cc;
    }
    float acc = 0.f;
    const int myPid = pid[row];
    for (int iS = 0; iS < nb; ++iS) {
      const int cc = bcol[iS];
      if (cc == row) continue;                 // self mask
      if (pid[cc] == myPid) continue;          // fake mask
      const float dd = sqrtf(fmaxf(best[iS], 0.f) + 1e-12f);
      if (dd < 1.0f) { const float h = 1.0f - dd; acc += h * h; }
    }
    if (acc != 0.f) atomicAdd(out, acc * (1.0f / (float)N_PTS));
  }
}

extern "C" void kernel_launch(void* const* d_in, const int* in_sizes, int n_in,
                              void* d_out, int out_size, void* d_ws, size_t ws_size,
                              hipStream_t stream) {
  const float* emb = (const float*)d_in[0];
  const int*   sig = (const int*)d_in[1];   // [2, E_SIG] flat
  const int*   rnd = (const int*)d_in[2];   // [2, E_RND] flat
  const int*   pid = (const int*)d_in[3];
  float* out = (float*)d_out;
  float* sq  = (float*)d_ws;                // N_PTS floats of scratch

  hipMemsetAsync(d_out, 0, (size_t)out_size * sizeof(float), stream);

  sq_kernel<<<(N_PTS + 255) / 256, 256, 0, stream>>>(emb, sq);
  edge_kernel<<<(E_SIG_N + 255) / 256, 256, 0, stream>>>(
      emb, sig, pid, out, E_SIG_N, 1.0f / (float)E_SIG_N, /*hinge=*/0);
  edge_kernel<<<(E_RND_N + 255) / 256, 256, 0, stream>>>(
      emb, rnd, pid, out, E_RND_N, 1.0f / (float)E_RND_N, /*hinge=*/1);
  knn_kernel<<<N_PTS / TILE, KNN_THREADS, 0, stream>>>(emb, sq, pid, out);
}